// GATNet_68822555951597
// MI455X (gfx1250) — compile-verified
//
#include <hip/hip_runtime.h>
#include <math.h>

typedef __attribute__((ext_vector_type(2))) float v2f;
typedef __attribute__((ext_vector_type(8))) float v8f;

#define NEG_SLOPE 0.2f
#define BN_EPS 1e-5f

// ---------------------------------------------------------------------------
// float atomic max via integer bit tricks (works for any sign, init = -inf)
// ---------------------------------------------------------------------------
__device__ inline void atomic_max_f32(float* addr, float v) {
  if (v >= 0.0f) {
    atomicMax((int*)addr, __float_as_int(v));
  } else {
    atomicMin((unsigned int*)addr, __float_as_uint(v));
  }
}

__global__ void fill_f32(float* __restrict__ p, float v, int n) {
  int i = blockIdx.x * blockDim.x + threadIdx.x;
  if (i < n) p[i] = v;
}

// ---------------------------------------------------------------------------
// f32 WMMA GEMM: C[M,N] = A[M,K] @ B[K,N], row-major, exact fp32 via
// V_WMMA_F32_16X16X4_F32. One wave -> 32(M)x64(N) output (2 M-tiles x 4
// N-tiles = 8 accumulators), so the 8 B-fragment loads per k-step feed 8
// WMMAs (10 loads / 8 wmma vs 9 / 4 before). Requires N%64==0, K%4==0,
// M%16==0; M%32 != 0 handled by clamping the 2nd M-tile (loads duplicated
// in-bounds, stores skipped; branch is wave-uniform so EXEC stays all-1s
// for WMMA as the ISA requires).
// Fragment layouts per CDNA5 ISA 7.12.2:
//   A 16x4 : lane<16: M=lane, VGPR0=K0,VGPR1=K1 ; lane>=16: K2,K3
//   B 4x16 : VGPR r holds row K=r (lanes 0-15) / K=r+2 (lanes 16-31), N=lane&15
//   C/D    : VGPR r holds M=r (lanes 0-15) / M=r+8 (lanes 16-31), N=lane&15
// ---------------------------------------------------------------------------
__global__ __launch_bounds__(256) void gemm_wmma_f32(
    const float* __restrict__ A, const float* __restrict__ B,
    float* __restrict__ C, int M, int N, int K, int tilesNw, int totalWaves)
{
  const int wave = blockIdx.x * 8 + (threadIdx.x >> 5);
  if (wave >= totalWaves) return;
  const int lane = threadIdx.x & 31;
  const int half = lane >> 4;     // 0: lanes 0-15, 1: lanes 16-31
  const int sub  = lane & 15;
  const int tN = wave % tilesNw;
  const int tM = wave / tilesNw;
  const int m0 = tM * 32;
  const int n0 = tN << 6;
  const bool m1ok = (m0 + 16) < M;
  const int m1 = m1ok ? (m0 + 16) : m0;   // clamp: duplicate loads, no store

  v8f acc0 = {}; v8f acc1 = {}; v8f acc2 = {}; v8f acc3 = {};
  v8f acc4 = {}; v8f acc5 = {}; v8f acc6 = {}; v8f acc7 = {};
  const float* Arow0 = A + (size_t)(m0 + sub) * K + 2 * half;
  const float* Arow1 = A + (size_t)(m1 + sub) * K + 2 * half;

  for (int k = 0; k < K; k += 4) {
    v2f a0; a0.x = Arow0[k]; a0.y = Arow0[k + 1];
    v2f a1; a1.x = Arow1[k]; a1.y = Arow1[k + 1];
    const float* B0 = B + (size_t)(k + 2 * half) * N + n0 + sub;
    const float* B1 = B0 + N;
    v2f b0; b0.x = B0[0];  b0.y = B1[0];
    v2f b1; b1.x = B0[16]; b1.y = B1[16];
    v2f b2; b2.x = B0[32]; b2.y = B1[32];
    v2f b3; b3.x = B0[48]; b3.y = B1[48];
    acc0 = __builtin_amdgcn_wmma_f32_16x16x4_f32(false, a0, false, b0, (short)0, acc0, false, false);
    acc1 = __builtin_amdgcn_wmma_f32_16x16x4_f32(false, a0, false, b1, (short)0, acc1, false, false);
    acc2 = __builtin_amdgcn_wmma_f32_16x16x4_f32(false, a0, false, b2, (short)0, acc2, false, false);
    acc3 = __builtin_amdgcn_wmma_f32_16x16x4_f32(false, a0, false, b3, (short)0, acc3, false, false);
    acc4 = __builtin_amdgcn_wmma_f32_16x16x4_f32(false, a1, false, b0, (short)0, acc4, false, false);
    acc5 = __builtin_amdgcn_wmma_f32_16x16x4_f32(false, a1, false, b1, (short)0, acc5, false, false);
    acc6 = __builtin_amdgcn_wmma_f32_16x16x4_f32(false, a1, false, b2, (short)0, acc6, false, false);
    acc7 = __builtin_amdgcn_wmma_f32_16x16x4_f32(false, a1, false, b3, (short)0, acc7, false, false);
  }

#pragma unroll
  for (int r = 0; r < 8; ++r) {
    float* Crow = C + (size_t)(m0 + r + 8 * half) * N + n0 + sub;
    Crow[0]  = acc0[r];
    Crow[16] = acc1[r];
    Crow[32] = acc2[r];
    Crow[48] = acc3[r];
  }
  if (m1ok) {
#pragma unroll
    for (int r = 0; r < 8; ++r) {
      float* Crow = C + (size_t)(m1 + r + 8 * half) * N + n0 + sub;
      Crow[0]  = acc4[r];
      Crow[16] = acc5[r];
      Crow[32] = acc6[r];
      Crow[48] = acc7[r];
    }
  }
}

// ---------------------------------------------------------------------------
// s_src[n,h] = sum_c h[n,h,c]*a_src[h,c] ; likewise s_dst. C = 128 fixed.
// One wave per (n,h); lane-strided coalesced loads + shuffle reduction.
// ---------------------------------------------------------------------------
__global__ __launch_bounds__(256) void attn_scores(
    const float* __restrict__ h, const float* __restrict__ a_src,
    const float* __restrict__ a_dst, float* __restrict__ s_src,
    float* __restrict__ s_dst, int N, int H)
{
  const int wave = blockIdx.x * 8 + (threadIdx.x >> 5);
  const int lane = threadIdx.x & 31;
  if (wave >= N * H) return;
  const int hh = wave % H;
  const float* row = h + (size_t)wave * 128;
  const float* as = a_src + hh * 128;
  const float* ad = a_dst + hh * 128;
  float ss = 0.f, sd = 0.f;
#pragma unroll
  for (int c = 0; c < 128; c += 32) {
    float v = row[c + lane];
    ss += v * as[c + lane];
    sd += v * ad[c + lane];
  }
#pragma unroll
  for (int off = 16; off > 0; off >>= 1) {
    ss += __shfl_down(ss, off, 32);
    sd += __shfl_down(sd, off, 32);
  }
  if (lane == 0) { s_src[wave] = ss; s_dst[wave] = sd; }
}

// pass 1: e = leaky_relu(s_src[src]+s_dst[dst]); segment max over dst
__global__ void edge_max_kernel(const int* __restrict__ src, const int* __restrict__ dst,
                                const float* __restrict__ s_src, const float* __restrict__ s_dst,
                                float* __restrict__ e_val, float* __restrict__ e_max,
                                int E, int H)
{
  int i = blockIdx.x * blockDim.x + threadIdx.x;
  if (i >= E * H) return;
  int e = i / H, hh = i - e * H;
  float v = s_src[src[e] * H + hh] + s_dst[dst[e] * H + hh];
  v = v > 0.f ? v : NEG_SLOPE * v;
  e_val[i] = v;
  atomic_max_f32(&e_max[dst[e] * H + hh], v);
}

// pass 2: e_exp = exp(e - max); segment sum -> denom
__global__ void edge_exp_kernel(const int* __restrict__ dst,
                                const float* __restrict__ e_val, const float* __restrict__ e_max,
                                float* __restrict__ e_exp, float* __restrict__ denom,
                                int E, int H)
{
  int i = blockIdx.x * blockDim.x + threadIdx.x;
  if (i >= E * H) return;
  int e = i / H, hh = i - e * H;
  float ex = expf(e_val[i] - e_max[dst[e] * H + hh]);
  e_exp[i] = ex;
  atomicAdd(&denom[dst[e] * H + hh], ex);
}

// pass 3: agg[dst,h,:] += h[src,h,:] * alpha. One wave per (e,h), float4 lanes.
__global__ __launch_bounds__(256) void edge_aggregate(
    const int* __restrict__ src, const int* __restrict__ dst,
    const float* __restrict__ h, const float* __restrict__ e_exp,
    const float* __restrict__ denom, float* __restrict__ agg, int E, int H)
{
  const int wave = blockIdx.x * 8 + (threadIdx.x >> 5);
  const int lane = threadIdx.x & 31;
  if (wave >= E * H) return;
  const int e = wave / H, hh = wave - e * H;
  const int s = src[e], d = dst[e];
  const float alpha = e_exp[wave] / (denom[d * H + hh] + 1e-16f);
  const float4 v = ((const float4*)(h + ((size_t)s * H + hh) * 128))[lane];
  float* o = agg + ((size_t)d * H + hh) * 128 + lane * 4;
  atomicAdd(o + 0, v.x * alpha);
  atomicAdd(o + 1, v.y * alpha);
  atomicAdd(o + 2, v.z * alpha);
  atomicAdd(o + 3, v.w * alpha);
}

// BN stats over N rows of 1024 cols; per-block register accumulation of
// 64 rows then atomicAdd partials (sums[0:1024]=sum, sums[1024:2048]=sumsq).
__global__ __launch_bounds__(256) void bn_stats(
    const float* __restrict__ agg, const float* __restrict__ b1,
    float* __restrict__ sums, int N)
{
  const int row0 = blockIdx.x * 64;
  const int col = threadIdx.x;
  float s0 = 0, s1 = 0, s2 = 0, s3 = 0, q0 = 0, q1 = 0, q2 = 0, q3 = 0;
  const float bv0 = b1[col], bv1 = b1[col + 256], bv2 = b1[col + 512], bv3 = b1[col + 768];
  const int rows = min(64, N - row0);
  for (int r = 0; r < rows; ++r) {
    const float* rp = agg + (size_t)(row0 + r) * 1024;
    float v0 = rp[col] + bv0, v1 = rp[col + 256] + bv1;
    float v2 = rp[col + 512] + bv2, v3 = rp[col + 768] + bv3;
    s0 += v0; q0 += v0 * v0; s1 += v1; q1 += v1 * v1;
    s2 += v2; q2 += v2 * v2; s3 += v3; q3 += v3 * v3;
  }
  atomicAdd(&sums[col], s0);        atomicAdd(&sums[1024 + col], q0);
  atomicAdd(&sums[col + 256], s1);  atomicAdd(&sums[1280 + col], q1);
  atomicAdd(&sums[col + 512], s2);  atomicAdd(&sums[1536 + col], q2);
  atomicAdd(&sums[col + 768], s3);  atomicAdd(&sums[1792 + col], q3);
}

// in-place: y = ELU(gamma*(v+b1-mu)*rsqrt(var+eps)+beta)
__global__ void bn_elu(float* __restrict__ agg, const float* __restrict__ b1,
                       const float* __restrict__ gamma, const float* __restrict__ beta,
                       const float* __restrict__ sums, int N)
{
  int idx = blockIdx.x * blockDim.x + threadIdx.x;
  if (idx >= N * 1024) return;
  int c = idx & 1023;
  float inv_n = 1.0f / (float)N;
  float mu = sums[c] * inv_n;
  float var = sums[1024 + c] * inv_n - mu * mu;
  float v = agg[idx] + b1[c];
  float y = gamma[c] * (v - mu) * rsqrtf(var + BN_EPS) + beta[c];
  agg[idx] = y > 0.f ? y : (expf(y) - 1.0f);
}

__global__ void add_bias_out(const float* __restrict__ agg2, const float* __restrict__ b2,
                             float* __restrict__ out, int N)
{
  int idx = blockIdx.x * blockDim.x + threadIdx.x;
  if (idx >= N * 128) return;
  out[idx] = agg2[idx] + b2[idx & 127];
}

// ---------------------------------------------------------------------------
extern "C" void kernel_launch(void* const* d_in, const int* in_sizes, int n_in,
                              void* d_out, int out_size, void* d_ws, size_t ws_size,
                              hipStream_t stream)
{
  const float* x      = (const float*)d_in[0];
  const int*   ei     = (const int*)d_in[1];   // [2,E] int32
  const float* W1     = (const float*)d_in[2];
  const float* a_src1 = (const float*)d_in[3];
  const float* a_dst1 = (const float*)d_in[4];
  const float* b1     = (const float*)d_in[5];
  const float* gamma  = (const float*)d_in[6];
  const float* beta   = (const float*)d_in[7];
  const float* W2     = (const float*)d_in[8];
  const float* a_src2 = (const float*)d_in[9];
  const float* a_dst2 = (const float*)d_in[10];
  const float* b2     = (const float*)d_in[11];
  float* out = (float*)d_out;

  const int N = in_sizes[0] / 128;
  const int E = in_sizes[1] / 2;
  const int* src = ei;
  const int* dst = ei + E;

  float* ws = (float*)d_ws;
  size_t off = 0;
  float* hbuf  = ws + off; off += (size_t)N * 1024;   // layer-1 h (pre-attn)
  float* agg1  = ws + off; off += (size_t)N * 1024;   // layer-1 aggregate / h1n
  float* ssrc1 = ws + off; off += (size_t)N * 8;
  float* sdst1 = ws + off; off += (size_t)N * 8;
  float* emax1 = ws + off; off += (size_t)N * 8;
  float* den1  = ws + off; off += (size_t)N * 8;
  float* eval1 = ws + off; off += (size_t)E * 8;
  float* eexp1 = ws + off; off += (size_t)E * 8;
  float* bnsum = ws + off; off += 2048;
  // layer-2 scratch aliases hbuf (dead after edge_aggregate of layer 1);
  // all writes below are issued after the last read of hbuf in stream order.
  float* h2    = hbuf;                       // N*128
  float* ssrc2 = hbuf + (size_t)N * 128;
  float* sdst2 = ssrc2 + N;
  float* emax2 = sdst2 + N;
  float* den2  = emax2 + N;
  float* eval2 = den2 + N;
  float* eexp2 = eval2 + E;
  float* agg2  = eexp2 + E;                  // N*128

  // ---- layer 1 --------------------------------------------------------
  hipMemsetAsync(agg1, 0, (size_t)N * 1024 * sizeof(float), stream);
  hipMemsetAsync(den1, 0, (size_t)N * 8 * sizeof(float), stream);
  hipMemsetAsync(bnsum, 0, 2048 * sizeof(float), stream);
  fill_f32<<<(N * 8 + 255) / 256, 256, 0, stream>>>(emax1, -INFINITY, N * 8);

  {
    int tilesNw = 1024 / 64, tilesM = (N + 31) / 32;
    int totalWaves = tilesNw * tilesM;
    gemm_wmma_f32<<<(totalWaves + 7) / 8, 256, 0, stream>>>(x, W1, hbuf, N, 1024, 128,
                                                            tilesNw, totalWaves);
  }
  attn_scores<<<(N * 8 + 7) / 8, 256, 0, stream>>>(hbuf, a_src1, a_dst1, ssrc1, sdst1, N, 8);
  edge_max_kernel<<<(E * 8 + 255) / 256, 256, 0, stream>>>(src, dst, ssrc1, sdst1,
                                                           eval1, emax1, E, 8);
  edge_exp_kernel<<<(E * 8 + 255) / 256, 256, 0, stream>>>(dst, eval1, emax1,
                                                           eexp1, den1, E, 8);
  edge_aggregate<<<(E * 8 + 7) / 8, 256, 0, stream>>>(src, dst, hbuf, eexp1, den1,
                                                      agg1, E, 8);   // last hbuf read
  bn_stats<<<(N + 63) / 64, 256, 0, stream>>>(agg1, b1, bnsum, N);
  bn_elu<<<(N * 1024 + 255) / 256, 256, 0, stream>>>(agg1, b1, gamma, beta, bnsum, N);

  // ---- layer 2 (scratch now safely aliases hbuf) ----------------------
  hipMemsetAsync(den2, 0, (size_t)N * sizeof(float), stream);
  hipMemsetAsync(agg2, 0, (size_t)N * 128 * sizeof(float), stream);
  fill_f32<<<(N + 255) / 256, 256, 0, stream>>>(emax2, -INFINITY, N);

  {
    int tilesNw = 128 / 64, tilesM = (N + 31) / 32;
    int totalWaves = tilesNw * tilesM;
    gemm_wmma_f32<<<(totalWaves + 7) / 8, 256, 0, stream>>>(agg1, W2, h2, N, 128, 1024,
                                                            tilesNw, totalWaves);
  }
  attn_scores<<<(N + 7) / 8, 256, 0, stream>>>(h2, a_src2, a_dst2, ssrc2, sdst2, N, 1);
  edge_max_kernel<<<(E + 255) / 256, 256, 0, stream>>>(src, dst, ssrc2, sdst2,
                                                       eval2, emax2, E, 1);
  edge_exp_kernel<<<(E + 255) / 256, 256, 0, stream>>>(dst, eval2, emax2,
                                                       eexp2, den2, E, 1);
  edge_aggregate<<<(E + 7) / 8, 256, 0, stream>>>(src, dst, h2, eexp2, den2, agg2, E, 1);

  add_bias_out<<<(N * 128 + 255) / 256, 256, 0, stream>>>(agg2, b2, out, N);
}